// WKVMemory_59536836657395
// MI455X (gfx1250) — compile-verified
//
#include <hip/hip_runtime.h>

// Problem constants (match reference: B=8, T=4096, D=1024, fp32 in/out).
#define BB 8
#define TT 4096
#define DD 1024
#define WW 2      // timesteps per tile
#define NBUF 4    // pipeline stages (3 tiles in flight + 1 being consumed)

typedef float v4f __attribute__((ext_vector_type(4)));

// Low 32 bits of a generic LDS pointer == LDS byte offset (flat LDS addr is
// {SHARED_BASE[63:32], offset[31:0]}), which is what the async VDST wants.
__device__ __forceinline__ unsigned lds_addr32(const void* p) {
  return (unsigned)(uintptr_t)p;
}

// CDNA5 async copy engine: per-lane b128 global -> LDS, tracked by ASYNCcnt.
__device__ __forceinline__ void async_ld_b128(unsigned ldsoff, const float* g) {
  asm volatile("global_load_async_to_lds_b128 %0, %1, off"
               :: "v"(ldsoff), "v"(g)
               : "memory");
}
// Last-use variant for the second (final) read of the inputs: free L2 on hit.
__device__ __forceinline__ void async_ld_b128_lu(unsigned ldsoff, const float* g) {
  asm volatile("global_load_async_to_lds_b128 %0, %1, off th:TH_LOAD_LU"
               :: "v"(ldsoff), "v"(g)
               : "memory");
}

#define WAIT_ASYNCCNT(n) asm volatile("s_wait_asynccnt " #n ::: "memory")

__device__ __forceinline__ float fast_rcp(float x) {
  float r = __builtin_amdgcn_rcpf(x);
  return r * (2.0f - x * r);   // one Newton step
}

// -----------------------------------------------------------------------------
// Pass 1 (WRITE_OUT=false): per-chunk local affine scan from zero; store carry.
// Pass 3 (WRITE_OUT=true) : re-scan seeded with exclusive carry; write outputs.
// Block = one (b, chunk). 256 threads x 4 channels = all of D. Per-wave async
// pipeline keeps 3 tiles (12 b128 asyncs, 6 KB) in flight; each lane consumes
// only the LDS it loaded itself, so ordering needs only s_wait_asynccnt.
// -----------------------------------------------------------------------------
template<bool WRITE_OUT>
__global__ __launch_bounds__(256)
void wkv_pass(const float* __restrict__ values,
              const float* __restrict__ imps,
              const float* __restrict__ log_gain,
              const float* __restrict__ log_decay,
              float* __restrict__ out,
              float* __restrict__ carryC,
              float* __restrict__ carryN,
              int C, int L, int use_carry) {
  // [NBUF][WW timesteps][2 arrays (v, imp)][D floats] = 64 KB
  __shared__ float smem[NBUF * WW * 2 * DD];

  // Pass 3 walks blocks in reverse so its first reads hit the tail of the
  // stream pass 1 left resident in the 192 MB L2.
  const unsigned bid = WRITE_OUT ? (gridDim.x - 1u - blockIdx.x) : blockIdx.x;
  const int b  = bid / C;
  const int c  = bid % C;
  const int d4 = threadIdx.x * 4;

  const size_t chunk = ((size_t)b * TT + (size_t)c * L) * DD;
  const float* vb = values + chunk;
  const float* ib = imps   + chunk;

  // Per-channel parameters.
  v4f ld = *(const v4f*)(log_decay + d4);
  v4f decay, gain = {0.f, 0.f, 0.f, 0.f};
#pragma unroll
  for (int i = 0; i < 4; ++i) decay[i] = expf(-expf(ld[i]));
  if constexpr (WRITE_OUT) {
    v4f lg = *(const v4f*)(log_gain + d4);
#pragma unroll
    for (int i = 0; i < 4; ++i) gain[i] = expf(lg[i]) - 1.0f;
  }

  const size_t cidx = (size_t)(c * BB + b) * DD + d4;
  v4f cont = {0.f, 0.f, 0.f, 0.f};
  v4f norm = {0.f, 0.f, 0.f, 0.f};
  if (WRITE_OUT && use_carry) {
    cont = *(const v4f*)(carryC + cidx);
    norm = *(const v4f*)(carryN + cidx);
  }

  auto issue = [&](int tile) {
    const int s = tile & (NBUF - 1);
#pragma unroll
    for (int w = 0; w < WW; ++w) {
      const size_t g   = (size_t)(tile * WW + w) * DD + d4;
      const int   base = ((s * WW + w) * 2) * DD + d4;
      if constexpr (WRITE_OUT) {
        async_ld_b128_lu(lds_addr32(&smem[base]),      vb + g);
        async_ld_b128_lu(lds_addr32(&smem[base + DD]), ib + g);
      } else {
        async_ld_b128(lds_addr32(&smem[base]),      vb + g);
        async_ld_b128(lds_addr32(&smem[base + DD]), ib + g);
      }
    }
  };

  const int nt = L / WW;   // tiles per chunk (>= 16 for C <= 128)
  issue(0);
  if (nt > 1) issue(1);
  if (nt > 2) issue(2);

  for (int k = 0; k < nt; ++k) {
    // Keep 3 tiles in flight; each tile = 4 async instrs (per wave), so
    // "tile k complete" == asynccnt <= 4 * (#tiles issued beyond k).
    if (k + 3 < nt)      { issue(k + 3); WAIT_ASYNCCNT(12); }
    else if (k + 2 < nt) { WAIT_ASYNCCNT(8); }
    else if (k + 1 < nt) { WAIT_ASYNCCNT(4); }
    else                 { WAIT_ASYNCCNT(0); }

    const int s = k & (NBUF - 1);
#pragma unroll
    for (int w = 0; w < WW; ++w) {
      const int base = ((s * WW + w) * 2) * DD + d4;
      v4f vv = *(const v4f*)&smem[base];
      v4f ii = *(const v4f*)&smem[base + DD];
      v4f iv = ii * vv;
      cont = cont * decay + iv;
      norm = norm * decay + ii;
      if constexpr (WRITE_OUT) {
        v4f num = cont + gain * iv;
        v4f den = norm + gain * ii;
        v4f o;
#pragma unroll
        for (int i = 0; i < 4; ++i) o[i] = num[i] * fast_rcp(den[i]);
        __builtin_nontemporal_store(
            o, (v4f*)(out + chunk + (size_t)(k * WW + w) * DD + d4));
      }
    }
  }

  if constexpr (!WRITE_OUT) {
    *(v4f*)(carryC + cidx) = cont;
    *(v4f*)(carryN + cidx) = norm;
  }
}

// -----------------------------------------------------------------------------
// Pass 2: exclusive cross-chunk scan per channel. One thread per 4 channels
// (2048 threads total); converts local chunk sums into chunk-entry states
// in place using decay^L = exp(-exp(log_decay) * L).
// -----------------------------------------------------------------------------
__global__ __launch_bounds__(256)
void wkv_midscan(float* __restrict__ carryC, float* __restrict__ carryN,
                 const float* __restrict__ log_decay, int C, int L) {
  const int tid = blockIdx.x * blockDim.x + threadIdx.x;  // 0 .. B*D/4-1
  const int d4  = (tid & (DD / 4 - 1)) * 4;
  const int b   = tid / (DD / 4);

  v4f ld = *(const v4f*)(log_decay + d4);
  v4f dL;
#pragma unroll
  for (int i = 0; i < 4; ++i) dL[i] = expf(-expf(ld[i]) * (float)L);

  v4f rc = {0.f, 0.f, 0.f, 0.f};
  v4f rn = {0.f, 0.f, 0.f, 0.f};
  for (int c = 0; c < C; ++c) {
    const size_t idx = (size_t)(c * BB + b) * DD + d4;
    v4f lc = *(const v4f*)(carryC + idx);
    v4f ln = *(const v4f*)(carryN + idx);
    *(v4f*)(carryC + idx) = rc;   // exclusive prefix (state entering chunk c)
    *(v4f*)(carryN + idx) = rn;
    rc = rc * dL + lc;
    rn = rn * dL + ln;
  }
}

// -----------------------------------------------------------------------------
// Host-side launcher
// -----------------------------------------------------------------------------
extern "C" void kernel_launch(void* const* d_in, const int* in_sizes, int n_in,
                              void* d_out, int out_size, void* d_ws, size_t ws_size,
                              hipStream_t stream) {
  const float* values    = (const float*)d_in[0];
  const float* imps      = (const float*)d_in[1];
  const float* log_gain  = (const float*)d_in[2];
  const float* log_decay = (const float*)d_in[3];
  float* out = (float*)d_out;

  // Largest chunk count whose carry arrays fit in workspace (8 MB at C=128).
  int C = 128;
  while (C > 1 && (size_t)2 * C * BB * DD * sizeof(float) > ws_size) C >>= 1;
  const int L = TT / C;

  float* carryC = (float*)d_ws;
  float* carryN = carryC + (size_t)C * BB * DD;
  const int use_carry = (C > 1) ? 1 : 0;

  if (use_carry) {
    hipLaunchKernelGGL((wkv_pass<false>), dim3(BB * C), dim3(256), 0, stream,
                       values, imps, log_gain, log_decay, (float*)nullptr,
                       carryC, carryN, C, L, 0);
    hipLaunchKernelGGL(wkv_midscan, dim3((BB * DD / 4) / 256), dim3(256), 0,
                       stream, carryC, carryN, log_decay, C, L);
  }
  hipLaunchKernelGGL((wkv_pass<true>), dim3(BB * C), dim3(256), 0, stream,
                     values, imps, log_gain, log_decay, out,
                     carryC, carryN, C, L, use_carry);
}